// STICKYKVCache_LayerWise_46943992545511
// MI455X (gfx1250) — compile-verified
//
#include <hip/hip_runtime.h>
#include <hip/hip_bf16.h>
#include <stdint.h>

typedef __attribute__((ext_vector_type(2))) float v2f;
typedef __attribute__((ext_vector_type(8))) float v8f;

namespace skv {
constexpr int H = 16, S = 4096, HD = 128;
constexpr int OMEGA = 32, SINK = 4;
constexpr int NUM_WIN = 123;               // scored windows
constexpr int TOT_WIN = 127;               // full prompt windows
constexpr int NCOLS   = TOT_WIN * OMEGA;   // 4064 columns (attn cols 4..4067)
constexpr int NSTRIPS = NCOLS / 16;        // 254 sixteen-wide strips
constexpr int KEEP = 8, EVICT = NUM_WIN - KEEP;     // 8 / 115
constexpr int SCORE_END = SINK + NUM_WIN * OMEGA;   // 3940
constexpr int TAIL = S - SCORE_END;                 // 156
constexpr int KEEP_TOK = SINK + KEEP * OMEGA + TAIL; // 416
}

using namespace skv;

// ---------------------------------------------------------------------------
// Kernel 1: column sums of attn_scores over all 4096 query rows, computed as
// ones[16x4] x B[4x16] WMMA accumulation. One wave per (head, 16-col strip).
// Bandwidth-bound: 1.07 GB streamed; WMMA does the reduction arithmetic.
// ---------------------------------------------------------------------------
__global__ __launch_bounds__(256) void colsum_wmma_kernel(
    const float* __restrict__ attn, float* __restrict__ obs) {
  const int wave = threadIdx.x >> 5;
  const int lane = threadIdx.x & 31;
  const int gs   = blockIdx.x * 8 + wave;     // 0 .. H*NSTRIPS-1 (4064)
  const int h    = gs / NSTRIPS;
  const int strip= gs % NSTRIPS;
  const int n    = lane & 15;                 // column within strip
  const int half = lane >> 4;                 // which row-pair this lane feeds

  // lane's base pointer: head h, column (4 + strip*16 + n)
  const float* __restrict__ A =
      attn + (size_t)h * S * S + (size_t)(SINK + strip * 16 + n);

  v8f acc = {};
  v2f ones; ones.x = 1.0f; ones.y = 1.0f;     // A-matrix = all ones (16x4)

#pragma unroll 8
  for (int q = 0; q < S; q += 4) {
    const float* p = A + (size_t)(q + 2 * half) * S;
    __builtin_prefetch(p + (size_t)16 * S, 0, 0);  // global_prefetch_b8, 16 rows ahead
    v2f b;
    b.x = p[0];          // row q   (+2*half)
    b.y = p[S];          // row q+1 (+2*half)
    // D = ones(16x4) x B(4x16) + C  -> per-column sums replicated over 16 rows
    acc = __builtin_amdgcn_wmma_f32_16x16x4_f32(
        /*neg_a=*/false, ones, /*neg_b=*/false, b,
        /*c_mod=*/(short)0, acc, /*reuse_a=*/false, /*reuse_b=*/false);
  }
  // C/D layout: VGPR0, lanes 0-15 = (M=0, N=lane). All rows identical.
  if (lane < 16) obs[h * NCOLS + strip * 16 + n] = acc[0];
}

// ---------------------------------------------------------------------------
// Kernel 2: window sums (groups of 32 columns). win_scores = fwin[:, :123].
// ---------------------------------------------------------------------------
__global__ void window_sum_kernel(const float* __restrict__ obs,
                                  float* __restrict__ win,
                                  float* __restrict__ fwin) {
  const int h = blockIdx.x;
  const int w = threadIdx.x;
  if (w < TOT_WIN) {
    const float* p = obs + h * NCOLS + w * OMEGA;
    float s = 0.0f;
#pragma unroll
    for (int i = 0; i < OMEGA; ++i) s += p[i];
    fwin[h * TOT_WIN + w] = s;
    if (w < NUM_WIN) win[h * NUM_WIN + w] = s;
  }
}

// ---------------------------------------------------------------------------
// Kernel 3: per-head stable ascending argsort of 123 scores via rank counting.
// Emits loser ids (ws), sorted survivor ids (ws), loser_scores (out).
// ---------------------------------------------------------------------------
__global__ void sort_kernel(const float* __restrict__ win,
                            int* __restrict__ loser_ids,
                            int* __restrict__ surv_ids,
                            float* __restrict__ loser_scores) {
  const int h = blockIdx.x;
  const int tid = threadIdx.x;
  __shared__ float s[NUM_WIN];
  __shared__ int   ord[NUM_WIN];
  if (tid < NUM_WIN) s[tid] = win[h * NUM_WIN + tid];
  __syncthreads();
  if (tid < NUM_WIN) {
    const float si = s[tid];
    int rank = 0;
    for (int j = 0; j < NUM_WIN; ++j) {
      const float sj = s[j];
      rank += (sj < si) || (sj == si && j < tid);   // stable ascending
    }
    ord[rank] = tid;
  }
  __syncthreads();
  if (tid < EVICT) {
    loser_ids[h * 128 + tid] = ord[tid];
    loser_scores[h * EVICT + tid] = s[ord[tid]];
  }
  if (tid == 0) {
    int sv[KEEP];
    for (int i = 0; i < KEEP; ++i) sv[i] = ord[EVICT + i];
    for (int i = 1; i < KEEP; ++i) {          // sort 8 ids ascending
      int key = sv[i], j = i - 1;
      while (j >= 0 && sv[j] > key) { sv[j + 1] = sv[j]; --j; }
      sv[j + 1] = key;
    }
    for (int i = 0; i < KEEP; ++i) surv_ids[h * KEEP + i] = sv[i];
  }
}

// ---------------------------------------------------------------------------
// Kernel 4: gather evicted windows (32x128) and asymmetric int8 quantize per
// (head, window). grid = (EVICT, H, 2) with z selecting K vs V.
// float4 loads/stores: 4 x b128 per thread each way.
// ---------------------------------------------------------------------------
__global__ __launch_bounds__(256) void quantize_kernel(
    const float* __restrict__ k, const float* __restrict__ v,
    const int* __restrict__ loser_ids,
    float* __restrict__ qk, float* __restrict__ qks, float* __restrict__ qkz,
    float* __restrict__ qv, float* __restrict__ qvs, float* __restrict__ qvz) {
  const int w = blockIdx.x, h = blockIdx.y, t = blockIdx.z;
  const int tid = threadIdx.x;
  const float* __restrict__ src = (t == 0) ? k : v;
  float* __restrict__ qout = (t == 0) ? qk : qv;
  float* __restrict__ sout = (t == 0) ? qks : qvs;
  float* __restrict__ zout = (t == 0) ? qkz : qvz;

  constexpr int N4 = (OMEGA * HD) / 4;        // 1024 float4 elements
  constexpr int C4 = HD / 4;                  // 32 float4 per row
  __shared__ float4 buf[N4];                  // 16 KB
  __shared__ float smin[256], smax[256];
  __shared__ float s_scale, s_zp;

  const int lid  = loser_ids[h * 128 + w];
  const int tok0 = lid * OMEGA + SINK;

  float mn = 3.4e38f, mx = -3.4e38f;
#pragma unroll
  for (int i = 0; i < N4 / 256; ++i) {
    const int idx = tid + i * 256;            // float4 index
    const int r = idx / C4, c4 = idx % C4;
    int tok = tok0 + r; tok = tok < (S - 1) ? tok : (S - 1);
    const float4 x =
        *(const float4*)(src + ((size_t)h * S + tok) * HD + c4 * 4);
    buf[idx] = x;
    mn = fminf(mn, fminf(fminf(x.x, x.y), fminf(x.z, x.w)));
    mx = fmaxf(mx, fmaxf(fmaxf(x.x, x.y), fmaxf(x.z, x.w)));
  }
  smin[tid] = mn; smax[tid] = mx;
  __syncthreads();
  for (int off = 128; off > 0; off >>= 1) {
    if (tid < off) {
      smin[tid] = fminf(smin[tid], smin[tid + off]);
      smax[tid] = fmaxf(smax[tid], smax[tid + off]);
    }
    __syncthreads();
  }
  if (tid == 0) {
    const float lo = smin[0], hi = smax[0];
    const float scale = fmaxf((hi - lo) / 255.0f, 1e-8f);
    const float zp = rintf(-128.0f - lo / scale);   // RNE matches jnp.round
    s_scale = scale; s_zp = zp;
    sout[h * EVICT + w] = scale;
    zout[h * EVICT + w] = zp;
  }
  __syncthreads();
  const float scale = s_scale, zp = s_zp;
  float4* __restrict__ dst =
      (float4*)(qout + ((size_t)h * EVICT + w) * (OMEGA * HD));
#pragma unroll
  for (int i = 0; i < N4 / 256; ++i) {
    const int idx = tid + i * 256;
    const float4 x = buf[idx];
    float4 o;
    float q;
    q = fminf(127.0f, fmaxf(-128.0f, rintf(x.x / scale) + zp)); o.x = (float)(int8_t)q;
    q = fminf(127.0f, fmaxf(-128.0f, rintf(x.y / scale) + zp)); o.y = (float)(int8_t)q;
    q = fminf(127.0f, fmaxf(-128.0f, rintf(x.z / scale) + zp)); o.z = (float)(int8_t)q;
    q = fminf(127.0f, fmaxf(-128.0f, rintf(x.w / scale) + zp)); o.w = (float)(int8_t)q;
    dst[idx] = o;
  }
}

// ---------------------------------------------------------------------------
// Kernel 5: gather the kept tokens [sink | survivors | tail] -> new_k/new_v.
// grid = (104, 16); each 32-lane group copies one 128-float token row of K
// and V as float4 (b128) transfers; 4 tokens per block.
// ---------------------------------------------------------------------------
__global__ __launch_bounds__(128) void gather_keep_kernel(
    const float* __restrict__ k, const float* __restrict__ v,
    const int* __restrict__ surv_ids,
    float* __restrict__ newk, float* __restrict__ newv) {
  const int t  = blockIdx.x * 4 + (threadIdx.x >> 5);   // kept-token index 0..415
  const int h  = blockIdx.y;
  const int c4 = threadIdx.x & 31;                      // float4 column 0..31
  int tok;
  if (t < SINK) {
    tok = t;
  } else if (t < SINK + KEEP * OMEGA) {
    const int s = t - SINK;
    tok = surv_ids[h * KEEP + (s >> 5)] * OMEGA + SINK + (s & (OMEGA - 1));
  } else {
    tok = SCORE_END + (t - (SINK + KEEP * OMEGA));
  }
  const size_t si = ((size_t)h * S + tok) * HD + c4 * 4;
  const size_t di = ((size_t)h * KEEP_TOK + t) * HD + c4 * 4;
  *(float4*)(newk + di) = *(const float4*)(k + si);
  *(float4*)(newv + di) = *(const float4*)(v + si);
}

// ---------------------------------------------------------------------------
extern "C" void kernel_launch(void* const* d_in, const int* in_sizes, int n_in,
                              void* d_out, int out_size, void* d_ws, size_t ws_size,
                              hipStream_t stream) {
  const float* k    = (const float*)d_in[0];
  const float* v    = (const float*)d_in[1];
  const float* attn = (const float*)d_in[2];
  float* out = (float*)d_out;

  // Output layout (flat, reference return order, fp32)
  const size_t n_new = (size_t)H * KEEP_TOK * HD;        // 851,968
  const size_t n_q   = (size_t)H * EVICT * OMEGA * HD;   // 7,536,640
  const size_t n_sw  = (size_t)H * EVICT;                // 1,840
  float* o_newk = out;
  float* o_newv = o_newk + n_new;
  float* o_qk   = o_newv + n_new;
  float* o_qks  = o_qk   + n_q;
  float* o_qkz  = o_qks  + n_sw;
  float* o_qv   = o_qkz  + n_sw;
  float* o_qvs  = o_qv   + n_q;
  float* o_qvz  = o_qvs  + n_sw;
  float* o_win  = o_qvz  + n_sw;                 // H*123
  float* o_fwin = o_win  + (size_t)H * NUM_WIN;  // H*127
  float* o_los  = o_fwin + (size_t)H * TOT_WIN;  // H*115

  // Workspace layout (~270 KB)
  float* ws_obs   = (float*)d_ws;                               // H*4064 f32
  int*   ws_loser = (int*)((char*)d_ws + sizeof(float) * H * NCOLS); // H*128
  int*   ws_surv  = ws_loser + H * 128;                         // H*8

  // 1) WMMA column reduction over all queries (dominant 1.07 GB stream)
  colsum_wmma_kernel<<<(H * NSTRIPS) / 8, 256, 0, stream>>>(attn, ws_obs);

  // 2) Window sums -> win_scores / full_win_scores
  window_sum_kernel<<<H, 128, 0, stream>>>(ws_obs, o_win, o_fwin);

  // 3) Stable argsort -> losers / survivors / loser_scores
  sort_kernel<<<H, 128, 0, stream>>>(o_win, ws_loser, ws_surv, o_los);

  // 4) Gather + int8 quantize evicted windows for K and V
  quantize_kernel<<<dim3(EVICT, H, 2), 256, 0, stream>>>(
      k, v, ws_loser, o_qk, o_qks, o_qkz, o_qv, o_qvs, o_qvz);

  // 5) Gather kept tokens -> new_k / new_v
  gather_keep_kernel<<<dim3((KEEP_TOK + 3) / 4, H), 128, 0, stream>>>(
      k, v, ws_surv, o_newk, o_newv);
}